// GAT_Encoder_54786602828343
// MI455X (gfx1250) — compile-verified
//
#include <hip/hip_runtime.h>

#define HIDDIM 128
#define NEG_SLOPE 0.2f
#define BN_EPS 1e-5f

typedef float v2f __attribute__((ext_vector_type(2)));
typedef float v8f __attribute__((ext_vector_type(8)));

// ---- ordered-uint encoding for float atomic max ----
__device__ __forceinline__ unsigned f2ord(float f) {
    unsigned u = __float_as_uint(f);
    return (u & 0x80000000u) ? ~u : (u | 0x80000000u);
}
__device__ __forceinline__ float ord2f(unsigned o) {
    return (o & 0x80000000u) ? __uint_as_float(o & 0x7FFFFFFFu) : __uint_as_float(~o);
}
// encode(-inf) == 0x007FFFFF
#define ORD_NEG_INF 0x007FFFFFu

// ------------------------------------------------------------------
// init: zero accumulator, denom, BN sums; set segment-max to -inf
// ------------------------------------------------------------------
__global__ void init_kernel(float* __restrict__ acc, unsigned* __restrict__ mbits,
                            float* __restrict__ denom, float* __restrict__ bnsum, int n) {
    int i = blockIdx.x * blockDim.x + threadIdx.x;
    int tot = n * HIDDIM;
    if (i < tot) acc[i] = 0.0f;
    if (i < n) { mbits[i] = ORD_NEG_INF; denom[i] = 0.0f; }
    if (i < 2 * HIDDIM) bnsum[i] = 0.0f;
}

// ------------------------------------------------------------------
// GEMM: H[N,128] = X[N,128] @ W[128,128], f32 WMMA 16x16x4.
// One wave computes a 16-row x 128-col strip (8 accum tiles).
// W cached in LDS, *row-pair interleaved*:
//   Wl[(k>>1)*256 + n*2 + (k&1)] = W[k][n]
// so a lane's B pair (K=ka, K=ka+1) is one contiguous float2 -> ds_load_b64.
// A frag (16x4 f32): lane<16 holds K=k,k+1 ; lane>=16 holds K=k+2,k+3.
// B frag (4x16 f32): v0 = rows K=k (lanes 0-15) / K=k+2 (lanes 16-31), v1 = K=k+1 / K=k+3.
// C/D (16x16 f32): VGPR r: row m0+r (lanes 0-15), m0+8+r (lanes 16-31), col = lane&15.
// ------------------------------------------------------------------
__global__ __launch_bounds__(128) void gemm_wmma_kernel(const float* __restrict__ X,
                                                        const float* __restrict__ W,
                                                        float* __restrict__ H, int n) {
    __shared__ float Wl[HIDDIM * HIDDIM];   // 64 KB, pair-interleaved layout
    int t = threadIdx.x;
    const float4* Wv = (const float4*)W;
    // 4096 float4s, 128 threads -> 32 each; scatter into interleaved layout
    for (int i = t; i < HIDDIM * HIDDIM / 4; i += 128) {
        int k = i >> 5;              // source row (128 floats = 32 float4)
        int c4 = (i & 31) * 4;       // source col of first element
        float4 w4 = Wv[i];
        int base = (k >> 1) * (2 * HIDDIM) + (k & 1);
        Wl[base + (c4 + 0) * 2] = w4.x;
        Wl[base + (c4 + 1) * 2] = w4.y;
        Wl[base + (c4 + 2) * 2] = w4.z;
        Wl[base + (c4 + 3) * 2] = w4.w;
    }
    __syncthreads();

    int wave = t >> 5, lane = t & 31;
    int strip = blockIdx.x * 4 + wave;
    int m0 = strip * 16;
    if (m0 >= n) return;

    int hl  = lane >> 4;      // 0: lanes 0-15, 1: lanes 16-31
    int l16 = lane & 15;

    v8f c[8];
#pragma unroll
    for (int j = 0; j < 8; ++j) c[j] = (v8f){0,0,0,0,0,0,0,0};

    int arow = m0 + l16; if (arow >= n) arow = n - 1;
    const float* xrow = X + (size_t)arow * HIDDIM;

    for (int k = 0; k < HIDDIM; k += 4) {
        int ka = k + hl * 2;                      // even
        v2f a; a.x = xrow[ka]; a.y = xrow[ka + 1];   // global_load_b64
        const float2* brow = (const float2*)&Wl[(ka >> 1) * (2 * HIDDIM)];
#pragma unroll
        for (int j = 0; j < 8; ++j) {
            float2 bp = brow[j * 16 + l16];          // ds_load_b64
            v2f b; b.x = bp.x; b.y = bp.y;
            c[j] = __builtin_amdgcn_wmma_f32_16x16x4_f32(
                       false, a, false, b, (short)0, c[j], false, false);
        }
    }
#pragma unroll
    for (int j = 0; j < 8; ++j) {
        int col = j * 16 + l16;
#pragma unroll
        for (int r = 0; r < 8; ++r) {
            int row = m0 + r + hl * 8;
            if (row < n) H[(size_t)row * HIDDIM + col] = c[j][r];
        }
    }
}

// ------------------------------------------------------------------
// alpha_src[n] = h[n] . a_src ; alpha_dst[n] = h[n] . a_dst
// one wave32 per node, float4 per lane, shuffle reduction
// ------------------------------------------------------------------
__global__ __launch_bounds__(256) void alpha_kernel(const float* __restrict__ H,
                                                    const float* __restrict__ a_src,
                                                    const float* __restrict__ a_dst,
                                                    float* __restrict__ as, float* __restrict__ ad,
                                                    int n) {
    int wave = threadIdx.x >> 5, lane = threadIdx.x & 31;
    int node = blockIdx.x * 8 + wave;
    if (node >= n) return;
    float4 h4 = ((const float4*)(H + (size_t)node * HIDDIM))[lane];
    float4 s4 = ((const float4*)a_src)[lane];
    float4 d4 = ((const float4*)a_dst)[lane];
    float s = h4.x * s4.x + h4.y * s4.y + h4.z * s4.z + h4.w * s4.w;
    float d = h4.x * d4.x + h4.y * d4.y + h4.z * d4.z + h4.w * d4.w;
#pragma unroll
    for (int off = 16; off > 0; off >>= 1) {
        s += __shfl_down(s, off, 32);
        d += __shfl_down(d, off, 32);
    }
    if (lane == 0) { as[node] = s; ad[node] = d; }
}

// ------------------------------------------------------------------
// pass 1 over edges (incl. self loops e>=E): e = lrelu(as[src]+ad[dst]);
// store e, atomic segment-max into mbits[dst]
// ------------------------------------------------------------------
__global__ void edge_max_kernel(const long long* __restrict__ ei,
                                const float* __restrict__ as, const float* __restrict__ ad,
                                float* __restrict__ ev, unsigned* __restrict__ mbits,
                                int E_, int n) {
    int e = blockIdx.x * blockDim.x + threadIdx.x;
    int total = E_ + n;
    if (e >= total) return;
    int s, d;
    if (e < E_) { s = (int)ei[e]; d = (int)ei[(size_t)E_ + e]; }
    else        { s = d = e - E_; }
    float v = as[s] + ad[d];
    v = (v > 0.0f) ? v : NEG_SLOPE * v;
    ev[e] = v;
    atomicMax(&mbits[d], f2ord(v));
}

// ------------------------------------------------------------------
// pass 2: ex = exp(e - m[dst]); denom[dst] += ex; acc[dst] += ex * h[src]
// one wave32 per edge, 4 features per lane
// ------------------------------------------------------------------
__global__ __launch_bounds__(256) void edge_acc_kernel(const long long* __restrict__ ei,
                                                       const float* __restrict__ ev,
                                                       const unsigned* __restrict__ mbits,
                                                       const float* __restrict__ H,
                                                       float* __restrict__ denom,
                                                       float* __restrict__ acc,
                                                       int E_, int n) {
    int lane = threadIdx.x & 31;
    int e = blockIdx.x * 8 + (threadIdx.x >> 5);
    int total = E_ + n;
    if (e >= total) return;
    int s, d;
    if (e < E_) { s = (int)ei[e]; d = (int)ei[(size_t)E_ + e]; }
    else        { s = d = e - E_; }
    float ex = __expf(ev[e] - ord2f(mbits[d]));
    if (lane == 0) atomicAdd(&denom[d], ex);
    float4 h4 = ((const float4*)(H + (size_t)s * HIDDIM))[lane];
    float* ap = acc + (size_t)d * HIDDIM + lane * 4;
    atomicAdd(ap + 0, ex * h4.x);
    atomicAdd(ap + 1, ex * h4.y);
    atomicAdd(ap + 2, ex * h4.z);
    atomicAdd(ap + 3, ex * h4.w);
}

// ------------------------------------------------------------------
// per node: y = acc/denom + bias (in place); partial BN sums -> atomics
// block = 128 threads (1 per feature), 64 nodes per block
// ------------------------------------------------------------------
__global__ __launch_bounds__(128) void node_finish_kernel(float* __restrict__ acc,
                                                          const float* __restrict__ denom,
                                                          const float* __restrict__ bias,
                                                          float* __restrict__ bnsum, int n) {
    int f = threadIdx.x;
    int start = blockIdx.x * 64;
    float b = bias[f];
    float s = 0.0f, s2 = 0.0f;
    for (int i = 0; i < 64; ++i) {
        int node = start + i;
        if (node >= n) break;
        float y = acc[(size_t)node * HIDDIM + f] / denom[node] + b;
        acc[(size_t)node * HIDDIM + f] = y;
        s += y; s2 += y * y;
    }
    atomicAdd(&bnsum[f], s);
    atomicAdd(&bnsum[HIDDIM + f], s2);
}

// ------------------------------------------------------------------
// batchnorm (biased var) + relu, streaming elementwise
// ------------------------------------------------------------------
__global__ void bn_relu_kernel(const float* __restrict__ y, const float* __restrict__ bnsum,
                               const float* __restrict__ gamma, const float* __restrict__ beta,
                               float* __restrict__ out, int n) {
    int idx = blockIdx.x * blockDim.x + threadIdx.x;
    int tot = n * HIDDIM;
    if (idx >= tot) return;
    int f = idx & (HIDDIM - 1);
    float invn = 1.0f / (float)n;
    float mean = bnsum[f] * invn;
    float var  = bnsum[HIDDIM + f] * invn - mean * mean;
    float v = gamma[f] * (y[idx] - mean) * rsqrtf(var + BN_EPS) + beta[f];
    out[idx] = (v > 0.0f) ? v : 0.0f;
}

// ------------------------------------------------------------------
extern "C" void kernel_launch(void* const* d_in, const int* in_sizes, int n_in,
                              void* d_out, int out_size, void* d_ws, size_t ws_size,
                              hipStream_t stream) {
    const float*     x  = (const float*)d_in[0];
    const long long* ei = (const long long*)d_in[1];

    int N_ = in_sizes[0] / HIDDIM;
    int E_ = in_sizes[1] / 2;
    int total_edges = E_ + N_;
    float* out = (float*)d_out;

    // workspace carve (all 4-byte types)
    float*    bufG   = (float*)d_ws;                    // [N,128] gemm output h
    float*    bufAcc = bufG + (size_t)N_ * HIDDIM;      // [N,128] accumulator / y
    float*    ev     = bufAcc + (size_t)N_ * HIDDIM;    // [E+N] edge logits
    float*    as     = ev + total_edges;                // [N]
    float*    ad     = as + N_;                         // [N]
    unsigned* mbits  = (unsigned*)(ad + N_);            // [N]
    float*    denom  = (float*)(mbits + N_);            // [N]
    float*    bnsum  = denom + N_;                      // [256] sum | sumsq

    int initGrid = (N_ * HIDDIM + 255) / 256;
    int gemmGrid = (N_ + 63) / 64;          // 4 waves/block * 16 rows
    int alphaGrid = (N_ + 7) / 8;
    int emaxGrid = (total_edges + 255) / 256;
    int eaccGrid = (total_edges + 7) / 8;
    int nodeGrid = (N_ + 63) / 64;
    int bnGrid = (N_ * HIDDIM + 255) / 256;

    for (int layer = 0; layer < 2; ++layer) {
        const float* Xin = (layer == 0) ? x : out;
        const float* Wm  = (const float*)d_in[layer == 0 ? 2 : 8];
        const float* aS  = (const float*)d_in[layer == 0 ? 3 : 9];
        const float* aD  = (const float*)d_in[layer == 0 ? 4 : 10];
        const float* bi  = (const float*)d_in[layer == 0 ? 5 : 11];
        const float* ga  = (const float*)d_in[layer == 0 ? 6 : 12];
        const float* be  = (const float*)d_in[layer == 0 ? 7 : 13];

        init_kernel<<<initGrid, 256, 0, stream>>>(bufAcc, mbits, denom, bnsum, N_);
        gemm_wmma_kernel<<<gemmGrid, 128, 0, stream>>>(Xin, Wm, bufG, N_);
        alpha_kernel<<<alphaGrid, 256, 0, stream>>>(bufG, aS, aD, as, ad, N_);
        edge_max_kernel<<<emaxGrid, 256, 0, stream>>>(ei, as, ad, ev, mbits, E_, N_);
        edge_acc_kernel<<<eaccGrid, 256, 0, stream>>>(ei, ev, mbits, bufG, denom, bufAcc, E_, N_);
        node_finish_kernel<<<nodeGrid, 128, 0, stream>>>(bufAcc, denom, bi, bnsum, N_);
        bn_relu_kernel<<<bnGrid, 256, 0, stream>>>(bufAcc, bnsum, ga, be, out, N_);
    }
}